// RNN_33603824123977
// MI455X (gfx1250) — compile-verified
//
#include <hip/hip_runtime.h>
#include <cstdint>

#define T_STEPS 30
#define HID     20
#define BLOCK   256
#define BPT     4   // batch elements per thread

// LDS layout (dword offsets):
//   W1 : [T][2][HID]  at 0            (1200 dwords)
//   b1 : [T][HID]     at 1200         (600)
//   W2 : [T][HID]     at 1800         (600)
//   b2 : [T]          at 2400         (30)
#define OFF_B1 (T_STEPS * 2 * HID)
#define OFF_W2 (T_STEPS * 3 * HID)
#define OFF_B2 (T_STEPS * 4 * HID)
#define W_TOTAL (OFF_B2 + T_STEPS + 2)   // pad to even

typedef float v2f __attribute__((ext_vector_type(2)));
typedef float v4f __attribute__((ext_vector_type(4)));

// tanh(z) = 1 - 2/(exp(2z)+1), exp(2z) = 2^(z * 2*log2(e))
__device__ __forceinline__ float fast_tanh(float z) {
    float e = __builtin_amdgcn_exp2f(z * 2.885390081777927f); // v_exp_f32
    float r = __builtin_amdgcn_rcpf(e + 1.0f);                // v_rcp_f32
    return __builtin_fmaf(-2.0f, r, 1.0f);
}

// CDNA5 async global -> LDS staging (ASYNCcnt path).
__device__ __forceinline__ void stage_async_b128(const float* __restrict__ src,
                                                 float* lds_dst, int n_dwords,
                                                 int tid) {
    unsigned lds_base = (unsigned)(uintptr_t)lds_dst; // low 32 bits = LDS offset
    for (int i = tid; i < n_dwords / 4; i += BLOCK) {
        unsigned lds_addr = lds_base + (unsigned)(i * 16);
        unsigned goff     = (unsigned)(i * 16);
        asm volatile("global_load_async_to_lds_b128 %0, %1, %2"
                     :: "v"(lds_addr), "v"(goff), "s"(src)
                     : "memory");
    }
}

__device__ __forceinline__ void stage_async_b32(const float* __restrict__ src,
                                                float* lds_dst, int n_dwords,
                                                int tid) {
    unsigned lds_base = (unsigned)(uintptr_t)lds_dst;
    for (int i = tid; i < n_dwords; i += BLOCK) {
        unsigned lds_addr = lds_base + (unsigned)(i * 4);
        unsigned goff     = (unsigned)(i * 4);
        asm volatile("global_load_async_to_lds_b32 %0, %1, %2"
                     :: "v"(lds_addr), "v"(goff), "s"(src)
                     : "memory");
    }
}

// One recurrent timestep for BPT batch elements; packed 2-wide over pairs of
// hidden units so layer1/layer2 FMAs select v_pk_fma_f32 (VOP3P).
__device__ __forceinline__ void rnn_step(const float* __restrict__ w, int t,
                                         const float xv[BPT], float o[BPT]) {
    const v4f* wa = (const v4f*)(w + t * 2 * HID);        // W1[t][0][:]
    const v4f* wb = (const v4f*)(w + t * 2 * HID + HID);  // W1[t][1][:]
    const v4f* wc = (const v4f*)(w + OFF_B1 + t * HID);   // b1[t][:]
    const v4f* wd = (const v4f*)(w + OFF_W2 + t * HID);   // W2[t][:]
    const float bb2 = w[OFF_B2 + t];

    v2f acc[BPT], xb[BPT], ob[BPT];
#pragma unroll
    for (int k = 0; k < BPT; ++k) {
        acc[k] = (v2f){bb2, 0.0f};
        xb[k]  = (v2f){xv[k], xv[k]};
        ob[k]  = (v2f){o[k], o[k]};
    }

    const v2f vzero = {0.0f, 0.0f};
#pragma unroll
    for (int c = 0; c < HID / 4; ++c) {
        v4f a4 = wa[c], b4 = wb[c], c4 = wc[c], d4 = wd[c];
#pragma unroll
        for (int h = 0; h < 2; ++h) {
            v2f aa = h ? __builtin_shufflevector(a4, a4, 2, 3)
                       : __builtin_shufflevector(a4, a4, 0, 1);
            v2f bb = h ? __builtin_shufflevector(b4, b4, 2, 3)
                       : __builtin_shufflevector(b4, b4, 0, 1);
            v2f cc = h ? __builtin_shufflevector(c4, c4, 2, 3)
                       : __builtin_shufflevector(c4, c4, 0, 1);
            v2f dd = h ? __builtin_shufflevector(d4, d4, 2, 3)
                       : __builtin_shufflevector(d4, d4, 0, 1);
#pragma unroll
            for (int k = 0; k < BPT; ++k) {
                v2f hh = __builtin_elementwise_fma(xb[k], aa, cc); // x*a + b1
                hh = __builtin_elementwise_fma(ob[k], bb, hh);     // + o*b
                hh = __builtin_elementwise_max(hh, vzero);         // relu
                acc[k] = __builtin_elementwise_fma(hh, dd, acc[k]); // h*W2
            }
        }
    }

#pragma unroll
    for (int k = 0; k < BPT; ++k)
        o[k] = fast_tanh(acc[k].x + acc[k].y);
}

template<bool CHECK>
__device__ __forceinline__ void rnn_body(const float* __restrict__ x,
                                         float* __restrict__ out,
                                         const float* w,
                                         long base, long Btot) {
    float o[BPT];
    const float* xp[BPT];
    float* op[BPT];
    bool m[BPT];
#pragma unroll
    for (int k = 0; k < BPT; ++k) {
        long bidx = base + (long)k * BLOCK;
        o[k]  = 0.0f;
        m[k]  = CHECK ? (bidx < Btot) : true;
        xp[k] = x + bidx * T_STEPS;
        op[k] = out + bidx * T_STEPS;
    }

    // Software-pipelined x loads: fetch next t-pair before computing current.
    v2f xcur[BPT];
#pragma unroll
    for (int k = 0; k < BPT; ++k)
        xcur[k] = (!CHECK || m[k]) ? *(const v2f*)(xp[k]) : (v2f){0.0f, 0.0f};

    for (int tt = 0; tt < T_STEPS; tt += 2) {   // t-pairs: b64 load/store
        int tpf = (tt + 2 < T_STEPS) ? tt + 2 : tt;  // clamped prefetch index
        v2f xnxt[BPT];
#pragma unroll
        for (int k = 0; k < BPT; ++k)
            xnxt[k] = (!CHECK || m[k]) ? *(const v2f*)(xp[k] + tpf)
                                       : (v2f){0.0f, 0.0f};

        float xv0[BPT], xv1[BPT];
        v2f ov[BPT];
#pragma unroll
        for (int k = 0; k < BPT; ++k) {
            xv0[k] = xcur[k].x;
            xv1[k] = xcur[k].y;
        }

        rnn_step(w, tt, xv0, o);
#pragma unroll
        for (int k = 0; k < BPT; ++k) ov[k].x = o[k];

        rnn_step(w, tt + 1, xv1, o);

#pragma unroll
        for (int k = 0; k < BPT; ++k) {
            ov[k].y = o[k];
            if (!CHECK || m[k]) *(v2f*)(op[k] + tt) = ov[k];
            xcur[k] = xnxt[k];
        }
    }
}

__global__ __launch_bounds__(BLOCK) void rnn_scan_kernel(
        const float* __restrict__ x,
        const float* __restrict__ W1, const float* __restrict__ b1,
        const float* __restrict__ W2, const float* __restrict__ b2,
        float* __restrict__ out, long Btot) {
    __shared__ __align__(16) float w[W_TOTAL];
    const int tid = threadIdx.x;

    // Stage all weights into LDS via CDNA5 async global->LDS loads.
    stage_async_b128(W1, w,          T_STEPS * 2 * HID, tid);
    stage_async_b128(b1, w + OFF_B1, T_STEPS * HID,     tid);
    stage_async_b128(W2, w + OFF_W2, T_STEPS * HID,     tid);
    stage_async_b32 (b2, w + OFF_B2, T_STEPS,           tid);
    asm volatile("s_wait_asynccnt 0" ::: "memory");
    __syncthreads();

    long base = (long)blockIdx.x * (BLOCK * BPT) + tid;
    if (base + (long)(BPT - 1) * BLOCK < Btot)
        rnn_body<false>(x, out, w, base, Btot);
    else
        rnn_body<true>(x, out, w, base, Btot);
}

extern "C" void kernel_launch(void* const* d_in, const int* in_sizes, int n_in,
                              void* d_out, int out_size, void* d_ws, size_t ws_size,
                              hipStream_t stream) {
    const float* x  = (const float*)d_in[0];
    const float* W1 = (const float*)d_in[1];
    const float* b1 = (const float*)d_in[2];
    const float* W2 = (const float*)d_in[3];
    const float* b2 = (const float*)d_in[4];
    float* out = (float*)d_out;

    long B = (long)in_sizes[0] / T_STEPS;
    long per_block = (long)BLOCK * BPT;
    int  grid = (int)((B + per_block - 1) / per_block);

    rnn_scan_kernel<<<grid, BLOCK, 0, stream>>>(x, W1, b1, W2, b2, out, B);
}